// GAEEncoder_58995670778277
// MI455X (gfx1250) — compile-verified
//
#include <hip/hip_runtime.h>

typedef __attribute__((ext_vector_type(2))) float v2f;
typedef __attribute__((ext_vector_type(8))) float v8f;

#define HDIM 256
#define SB_STRIDE 72   // LDS row stride (floats): 2*72 mod 64 == 16 -> halves hit disjoint banks

// ---------------------------------------------------------------------------
// Tiled FP32 WMMA GEMM:  C[M,N] = A[M,K] @ B[K,N]  (+bias) (+ReLU)
// Block = 256 threads = 8 waves; block tile 128 rows x 64 cols.
// B slice (K x 64) staged once in LDS (shared by all 8 waves), A streamed from
// global. Each wave: one A fragment (16x4 f32) x four B fragments per K-step
// -> 4 v_wmma_f32_16x16x4_f32.
// A-frag layout (ISA 7.12.2, 32-bit A 16x4): lanes0-15 -> K=0,1 ; lanes16-31 -> K=2,3
// B-frag mirrors; C/D: VGPR r: lanes0-15 M=r, lanes16-31 M=r+8, N=lane&15.
// ---------------------------------------------------------------------------
__global__ __launch_bounds__(256)
void gemm_f32_wmma(const float* __restrict__ A, const float* __restrict__ B,
                   const float* __restrict__ bias, float* __restrict__ C,
                   int M, int K, int N, int applyRelu)
{
    __shared__ float sB[256 * SB_STRIDE];   // up to K=256 rows x 64 cols (padded)

    const int tid  = threadIdx.x;
    const int lane = tid & 31;
    const int wave = tid >> 5;
    const int half = lane >> 4;    // 0: lanes 0-15, 1: lanes 16-31
    const int l16  = lane & 15;

    const int row0 = blockIdx.y * 128 + wave * 16;
    const int col0 = blockIdx.x * 64;

    // ---- cooperative stage of B[:, col0:col0+64) into LDS ----
    {
        const int r = tid >> 4;            // 0..15
        const int c = (tid & 15) << 2;     // 0,4,...,60
        for (int kb = 0; kb < K; kb += 16) {
            const float4 v = *(const float4*)(B + (size_t)(kb + r) * N + col0 + c);
            *(float4*)(&sB[(kb + r) * SB_STRIDE + c]) = v;   // ds_store_b128
        }
    }
    __syncthreads();

    int aRow = row0 + l16;
    if (aRow > M - 1) aRow = M - 1;        // clamp: loads stay in-bounds, EXEC all-ones
    const float* Aptr = A + (size_t)aRow * K + half * 2;

    v8f acc[4];
    acc[0] = (v8f)0.0f; acc[1] = (v8f)0.0f; acc[2] = (v8f)0.0f; acc[3] = (v8f)0.0f;

    for (int kk = 0; kk < K; kk += 4) {
        const v2f a = *(const v2f*)(Aptr + kk);            // global_load_b64
        __builtin_prefetch(Aptr + kk + 32, 0, 0);          // global_prefetch_b8

        const float* s0 = &sB[(kk + half * 2) * SB_STRIDE + l16];
#pragma unroll
        for (int t = 0; t < 4; ++t) {
            v2f b;
            b.x = s0[t * 16];                               // row kk + half*2
            b.y = s0[SB_STRIDE + t * 16];                   // row kk + half*2 + 1
            acc[t] = __builtin_amdgcn_wmma_f32_16x16x4_f32(
                false, a, false, b, (short)0, acc[t], false, false);
        }
    }

#pragma unroll
    for (int t = 0; t < 4; ++t) {
        const int col = col0 + t * 16 + l16;
        const float bv = bias ? bias[col] : 0.0f;
#pragma unroll
        for (int r = 0; r < 8; ++r) {
            const int row = row0 + half * 8 + r;
            if (row < M) {
                float v = acc[t][r] + bv;
                if (applyRelu) v = v > 0.0f ? v : 0.0f;
                C[(size_t)row * N + col] = v;
            }
        }
    }
}

// ---------------------------------------------------------------------------
// Degree / normalization helpers
// ---------------------------------------------------------------------------
__global__ void deg_init(float* __restrict__ deg, int N)
{
    int i = blockIdx.x * blockDim.x + threadIdx.x;
    if (i < N) deg[i] = 1.0f;                      // self loop
}

__global__ void deg_edges(const int* __restrict__ dst, float* __restrict__ deg, int E)
{
    int e = blockIdx.x * blockDim.x + threadIdx.x;
    if (e < E) atomicAdd(&deg[dst[e]], 1.0f);
}

__global__ void deg_rsqrt(float* __restrict__ deg, int N)
{
    int i = blockIdx.x * blockDim.x + threadIdx.x;
    if (i < N) {
        float d = deg[i];
        deg[i] = d > 0.0f ? rsqrtf(d) : 0.0f;      // in place: deg -> dinv
    }
}

// agg[i][f] = m[i][f] * dinv[i]^2      (self-loop term; also initializes agg)
__global__ void gcn_self(const float* __restrict__ m, const float* __restrict__ dinv,
                         float* __restrict__ agg, long total)
{
    long idx = (long)blockIdx.x * blockDim.x + threadIdx.x;
    if (idx >= total) return;
    const int i = (int)(idx >> 8);                 // H == 256
    const float w = dinv[i] * dinv[i];
    agg[idx] = m[idx] * w;
}

// agg[dst] += m[src] * dinv[src]*dinv[dst]   (one thread per edge x 4-float chunk)
__global__ void gcn_scatter(const float* __restrict__ m, const int* __restrict__ src,
                            const int* __restrict__ dst, const float* __restrict__ dinv,
                            float* __restrict__ agg, int E)
{
    long gid = (long)blockIdx.x * blockDim.x + threadIdx.x;
    if (gid >= (long)E * 64) return;
    const int e = (int)(gid >> 6);
    const int c = (int)(gid & 63) << 2;
    const int s = src[e];
    const int d = dst[e];
    const float w = dinv[s] * dinv[d];
    const float4 v = *(const float4*)(m + (size_t)s * HDIM + c);
    float* o = agg + (size_t)d * HDIM + c;
    atomicAdd(o + 0, v.x * w);
    atomicAdd(o + 1, v.y * w);
    atomicAdd(o + 2, v.z * w);
    atomicAdd(o + 3, v.w * w);
}

// h[i][f] = relu(agg[i][f] + b[f])
__global__ void bias_relu(const float* __restrict__ agg, const float* __restrict__ b,
                          float* __restrict__ h, long total)
{
    long idx = (long)blockIdx.x * blockDim.x + threadIdx.x;
    if (idx >= total) return;
    const float v = agg[idx] + b[idx & (HDIM - 1)];
    h[idx] = v > 0.0f ? v : 0.0f;
}

// ---------------------------------------------------------------------------
extern "C" void kernel_launch(void* const* d_in, const int* in_sizes, int n_in,
                              void* d_out, int out_size, void* d_ws, size_t ws_size,
                              hipStream_t stream)
{
    const float* x      = (const float*)d_in[0];
    const int*   eidx   = (const int*)  d_in[1];
    const float* enc_w1 = (const float*)d_in[2];
    const float* enc_b1 = (const float*)d_in[3];
    const float* enc_w2 = (const float*)d_in[4];
    const float* enc_b2 = (const float*)d_in[5];
    const float* gcn_w  = (const float*)d_in[6];
    const float* gcn_b  = (const float*)d_in[7];

    const int F_IN = 128;
    const int N    = in_sizes[0] / F_IN;
    const int E    = in_sizes[1] / 2;
    const int L    = in_sizes[6] / (HDIM * HDIM);

    const int* src = eidx;       // edge_index[0]
    const int* dst = eidx + E;   // edge_index[1]

    float* h    = (float*)d_out;                     // N*H : persistent node features
    float* m    = (float*)d_ws;                      // N*H : h1 / per-layer messages
    float* agg  = m + (size_t)N * HDIM;              // N*H : scatter accumulator
    float* dinv = agg + (size_t)N * HDIM;            // N   : deg -> dinv (in place)

    const dim3 blk(256);
    const dim3 gg(HDIM / 64, (N + 127) / 128);       // GEMM grid: 4 x 782
    const int tb = 256;
    const long total = (long)N * HDIM;

    // Encoder: h1 = relu(x@W1 + b1); h = h1@W2 + b2
    gemm_f32_wmma<<<gg, blk, 0, stream>>>(x, enc_w1, enc_b1, m, N, F_IN, HDIM, 1);
    gemm_f32_wmma<<<gg, blk, 0, stream>>>(m, enc_w2, enc_b2, h, N, HDIM, HDIM, 0);

    // GCN normalization: deg (with self loop) -> dinv = rsqrt(deg)
    deg_init <<<(N + tb - 1) / tb, tb, 0, stream>>>(dinv, N);
    deg_edges<<<(E + tb - 1) / tb, tb, 0, stream>>>(dst, dinv, E);
    deg_rsqrt<<<(N + tb - 1) / tb, tb, 0, stream>>>(dinv, N);

    for (int l = 0; l < L; ++l) {
        gemm_f32_wmma<<<gg, blk, 0, stream>>>(h, gcn_w + (size_t)l * HDIM * HDIM,
                                              nullptr, m, N, HDIM, HDIM, 0);
        gcn_self<<<(int)((total + tb - 1) / tb), tb, 0, stream>>>(m, dinv, agg, total);
        const long st = (long)E * 64;
        gcn_scatter<<<(int)((st + tb - 1) / tb), tb, 0, stream>>>(m, src, dst, dinv, agg, E);
        bias_relu<<<(int)((total + tb - 1) / tb), tb, 0, stream>>>(
            agg, gcn_b + (size_t)l * HDIM, h, total);
    }
}